// KALayer_84232898609359
// MI455X (gfx1250) — compile-verified
//
#include <hip/hip_runtime.h>

typedef __attribute__((ext_vector_type(2))) float v2f;
typedef __attribute__((ext_vector_type(8))) float v8f;

#define IN_DIM  256
#define OUT_DIM 256
#define BATCH   256
#define NB      8            // GRID_N + K = 5 + 3 basis functions
#define KDEG    3
#define NKNOT   12           // (GRID_N+1) + 2*K extended knots
#define FDIM    9            // 1 (silu base) + NB spline features
#define KTOT    (IN_DIM * FDIM)   // 2304 = GEMM K dimension

// ---------------------------------------------------------------------------
// Phase 1: per (batch b, input feature in) build the 9-wide feature vector
//   F[b][in*9 + 0]     = silu(x[b,in])
//   F[b][in*9 + 1 + n] = B_n(x[b,in])   (cubic Cox-de Boor, 8 basis fns)
// Grid row is per input feature (rows of `grid` are tiled identical).
// ---------------------------------------------------------------------------
__global__ void __launch_bounds__(256) build_F(const float* __restrict__ x,
                                               const float* __restrict__ grid,
                                               float* __restrict__ F) {
  int tid = blockIdx.x * blockDim.x + threadIdx.x;   // 0 .. 65535
  int b   = tid >> 8;
  int in  = tid & 255;

  float xv = x[b * IN_DIM + in];

  // Extended knot vector t[0..11]
  float g[6];
#pragma unroll
  for (int j = 0; j < 6; ++j) g[j] = grid[in * 6 + j];
  float h = (g[5] - g[0]) * 0.2f;                    // (g_last - g_first)/GRID_N

  float t[NKNOT];
#pragma unroll
  for (int j = 0; j < 3; ++j) t[j] = g[0] - h * (float)(3 - j);
#pragma unroll
  for (int j = 0; j < 6; ++j) t[3 + j] = g[j];
#pragma unroll
  for (int j = 0; j < 3; ++j) t[9 + j] = g[5] + h * (float)(j + 1);

  // Degree-0 indicator basis (11 intervals)
  float B[11];
#pragma unroll
  for (int i = 0; i < 11; ++i)
    B[i] = (xv >= t[i] && xv < t[i + 1]) ? 1.0f : 0.0f;

  // Cox-de Boor elevation to degree 3 (matches reference recurrence exactly)
#pragma unroll
  for (int j = 1; j <= KDEG; ++j) {
#pragma unroll
    for (int i = 0; i < 11 - 1; ++i) {
      if (i < 11 - j) {
        B[i] = (xv - t[i])        / (t[i + j]     - t[i])     * B[i]
             + (t[i + j + 1] - xv) / (t[i + j + 1] - t[i + 1]) * B[i + 1];
      }
    }
  }

  float silu = xv / (1.0f + __expf(-xv));

  float* dst = F + b * KTOT + in * FDIM;
  dst[0] = silu;
#pragma unroll
  for (int n = 0; n < NB; ++n) dst[1 + n] = B[n];
}

// ---------------------------------------------------------------------------
// Phase 2: fold scales into a 256 x 2304 weight matrix
//   W[o][in*9 + 0]     = scale_base[s]
//   W[o][in*9 + 1 + n] = scale_spline[s] * coef[s][n],   s = o*256 + in
// ---------------------------------------------------------------------------
__global__ void __launch_bounds__(256) build_W(const float* __restrict__ coef,
                                               const float* __restrict__ sb,
                                               const float* __restrict__ ss,
                                               float* __restrict__ W) {
  int s  = blockIdx.x * blockDim.x + threadIdx.x;    // 0 .. 65535
  int o  = s >> 8;
  int in = s & 255;

  float wb = sb[s];
  float ws = ss[s];

  float* dst = W + o * KTOT + in * FDIM;
  dst[0] = wb;
#pragma unroll
  for (int n = 0; n < NB; ++n) dst[1 + n] = ws * coef[s * NB + n];
}

// ---------------------------------------------------------------------------
// Phase 3: y[256x256] = F[256x2304] * W^T  via V_WMMA_F32_16X16X4_F32.
// One 16x16 output tile per wave; 256 tiles = 32 blocks x 8 waves.
// A-load per lane:  F[bm*16 + (lane&15)][k + 2*(lane>>4) + {0,1}]   (float2)
// B-load per lane:  W[bo*16 + (lane&15)][k + 2*(lane>>4) + {0,1}]   (float2)
// Two accumulators break the serial WMMA C->D dependence chain.
// ---------------------------------------------------------------------------
__global__ void __launch_bounds__(256) kan_gemm(const float* __restrict__ F,
                                                const float* __restrict__ W,
                                                float* __restrict__ y) {
  int wave = threadIdx.x >> 5;
  int lane = threadIdx.x & 31;
  int tile = blockIdx.x * 8 + wave;                  // 0 .. 255
  int bm   = tile >> 4;                              // batch tile (M)
  int bo   = tile & 15;                              // output tile (N)

  int hi  = lane >> 4;                               // 0: K pair {0,1}, 1: {2,3}
  int l15 = lane & 15;

  const float* Arow = F + (size_t)(bm * 16 + l15) * KTOT + 2 * hi;
  const float* Brow = W + (size_t)(bo * 16 + l15) * KTOT + 2 * hi;

  v8f acc0 = {};
  v8f acc1 = {};

#pragma unroll 4
  for (int k = 0; k < KTOT; k += 8) {
    v2f a0 = *(const v2f*)(Arow + k);
    v2f b0 = *(const v2f*)(Brow + k);
    acc0 = __builtin_amdgcn_wmma_f32_16x16x4_f32(false, a0, false, b0,
                                                 (short)0, acc0, false, false);
    v2f a1 = *(const v2f*)(Arow + k + 4);
    v2f b1 = *(const v2f*)(Brow + k + 4);
    acc1 = __builtin_amdgcn_wmma_f32_16x16x4_f32(false, a1, false, b1,
                                                 (short)0, acc1, false, false);
  }

  v8f acc = acc0 + acc1;

  // C/D layout: acc[r] holds (M = r + 8*hi, N = l15)
#pragma unroll
  for (int r = 0; r < 8; ++r) {
    int row = bm * 16 + r + 8 * hi;                  // batch index
    int col = bo * 16 + l15;                         // output index
    y[row * OUT_DIM + col] = acc[r];
  }
}

// ---------------------------------------------------------------------------
extern "C" void kernel_launch(void* const* d_in, const int* in_sizes, int n_in,
                              void* d_out, int out_size, void* d_ws, size_t ws_size,
                              hipStream_t stream) {
  const float* x    = (const float*)d_in[0];
  const float* grid = (const float*)d_in[1];
  const float* coef = (const float*)d_in[2];
  const float* sb   = (const float*)d_in[3];
  const float* ss   = (const float*)d_in[4];

  float* F = (float*)d_ws;                 // 256 x 2304 fp32  (2.36 MB)
  float* W = F + (size_t)BATCH * KTOT;     // 256 x 2304 fp32  (2.36 MB)
  float* y = (float*)d_out;                // 256 x 256 fp32

  build_F<<<BATCH * IN_DIM / 256, 256, 0, stream>>>(x, grid, F);
  build_W<<<OUT_DIM * IN_DIM / 256, 256, 0, stream>>>(coef, sb, ss, W);
  kan_gemm<<<32, 256, 0, stream>>>(F, W, y);
}